// ColorConstancyLoss_56092272886151
// MI455X (gfx1250) — compile-verified
//
#include <hip/hip_runtime.h>
#include <math.h>

typedef float v2f __attribute__((ext_vector_type(2)));
typedef float v8f __attribute__((ext_vector_type(8)));

namespace {
constexpr int   kBins         = 64;
constexpr float kEps          = 1e-8f;
constexpr int   kB            = 16;
constexpr int   kN            = 512 * 512;          // pixels per image per channel
constexpr int   kBlocksPerImg = 64;
constexpr int   kPxPerBlock   = kN / kBlocksPerImg; // 4096
constexpr int   kThreads      = 256;
constexpr int   kP1Iters      = kPxPerBlock / (kThreads * 2); // 8 (2 px/lane/iter)
constexpr int   kP2Iters      = kPxPerBlock / kThreads;       // 16

// workspace layout in 32-bit words
constexpr int WS_XSUM  = 0;               // 16*3 floats
constexpr int WS_YSUM  = 48;              // 16*3 floats
constexpr int WS_XMIN  = 96;              // 16 ordered-encoded u32
constexpr int WS_XMAX  = 112;
constexpr int WS_YMIN  = 128;
constexpr int WS_YMAX  = 144;
constexpr int WS_XHIST = 160;             // 16*64 u32 counts
constexpr int WS_YHIST = WS_XHIST + kB * kBins;
constexpr int WS_WORDS = WS_YHIST + kB * kBins;   // 2208 words (~8.8 KB)
} // namespace

// Monotone float <-> uint encoding so uint atomicMin/Max implement float min/max.
__device__ __forceinline__ unsigned enc_f(float f) {
  unsigned u = __float_as_uint(f);
  return (u & 0x80000000u) ? ~u : (u | 0x80000000u);
}
__device__ __forceinline__ float dec_f(unsigned u) {
  u = (u & 0x80000000u) ? (u ^ 0x80000000u) : ~u;
  return __uint_as_float(u);
}

// Exact left-to-right luminance, no FMA contraction (bin index is trunc-sensitive).
__device__ __forceinline__ float grayf(float r, float g, float b) {
  float t = __fmul_rn(0.299f, r);
  t = __fadd_rn(t, __fmul_rn(0.587f, g));
  t = __fadd_rn(t, __fmul_rn(0.114f, b));
  return t;
}

__global__ void cc_init(unsigned* __restrict__ ws) {
  int i = blockIdx.x * blockDim.x + threadIdx.x;
  if (i >= WS_WORDS) return;
  unsigned v = 0u;
  if ((i >= WS_XMIN && i < WS_XMAX) || (i >= WS_YMIN && i < WS_YMAX)) v = 0xFFFFFFFFu; // +inf key
  ws[i] = v;
}

// Pass 1: channel sums via V_WMMA_F32_16X16X4_F32 (A = data, B = ones) + gray min/max.
__global__ __launch_bounds__(kThreads) void cc_pass1(const float* __restrict__ x,
                                                     const float* __restrict__ y,
                                                     unsigned* __restrict__ ws) {
  const int blk = blockIdx.x;
  const int b   = blk / kBlocksPerImg;
  const int s   = blk % kBlocksPerImg;
  const int tid = threadIdx.x;

  const float* xr = x + (size_t)b * 3 * kN;
  const float* xg = xr + kN;
  const float* xb = xg + kN;
  const float* yr = y + (size_t)b * 3 * kN;
  const float* yg = yr + kN;
  const float* yb = yg + kN;

  v8f c0 = {}, c1 = {}, c2 = {}, c3 = {}, c4 = {}, c5 = {};
  v2f ones; ones[0] = 1.0f; ones[1] = 1.0f;

  float xmn = __builtin_inff(),  xmx = -__builtin_inff();
  float ymn = __builtin_inff(),  ymx = -__builtin_inff();

  const int base = s * kPxPerBlock + tid * 2;
#pragma unroll
  for (int it = 0; it < kP1Iters; ++it) {
    const int p = base + it * (kThreads * 2);
    v2f a0 = *(const v2f*)(xr + p);
    v2f a1 = *(const v2f*)(xg + p);
    v2f a2 = *(const v2f*)(xb + p);
    v2f a3 = *(const v2f*)(yr + p);
    v2f a4 = *(const v2f*)(yg + p);
    v2f a5 = *(const v2f*)(yb + p);

    // D = A x ones + C : accumulates row-sums of the 16x4 A tile into C.
    c0 = __builtin_amdgcn_wmma_f32_16x16x4_f32(false, a0, false, ones, (short)0, c0, false, false);
    c1 = __builtin_amdgcn_wmma_f32_16x16x4_f32(false, a1, false, ones, (short)0, c1, false, false);
    c2 = __builtin_amdgcn_wmma_f32_16x16x4_f32(false, a2, false, ones, (short)0, c2, false, false);
    c3 = __builtin_amdgcn_wmma_f32_16x16x4_f32(false, a3, false, ones, (short)0, c3, false, false);
    c4 = __builtin_amdgcn_wmma_f32_16x16x4_f32(false, a4, false, ones, (short)0, c4, false, false);
    c5 = __builtin_amdgcn_wmma_f32_16x16x4_f32(false, a5, false, ones, (short)0, c5, false, false);

#pragma unroll
    for (int j = 0; j < 2; ++j) {
      float gx = grayf(a0[j], a1[j], a2[j]);
      float gy = grayf(a3[j], a4[j], a5[j]);
      xmn = fminf(xmn, gx); xmx = fmaxf(xmx, gx);
      ymn = fminf(ymn, gy); ymx = fmaxf(ymx, gy);
    }
  }

  // Per-lane: sum 8 D VGPRs (= 8 row-sums); + partner lane (xor 16) = full tile sum.
  float sum[6];
  {
    float t;
    t = c0[0]+c0[1]+c0[2]+c0[3]+c0[4]+c0[5]+c0[6]+c0[7]; t += __shfl_xor(t, 16); sum[0] = t;
    t = c1[0]+c1[1]+c1[2]+c1[3]+c1[4]+c1[5]+c1[6]+c1[7]; t += __shfl_xor(t, 16); sum[1] = t;
    t = c2[0]+c2[1]+c2[2]+c2[3]+c2[4]+c2[5]+c2[6]+c2[7]; t += __shfl_xor(t, 16); sum[2] = t;
    t = c3[0]+c3[1]+c3[2]+c3[3]+c3[4]+c3[5]+c3[6]+c3[7]; t += __shfl_xor(t, 16); sum[3] = t;
    t = c4[0]+c4[1]+c4[2]+c4[3]+c4[4]+c4[5]+c4[6]+c4[7]; t += __shfl_xor(t, 16); sum[4] = t;
    t = c5[0]+c5[1]+c5[2]+c5[3]+c5[4]+c5[5]+c5[6]+c5[7]; t += __shfl_xor(t, 16); sum[5] = t;
  }
#pragma unroll
  for (int o = 16; o > 0; o >>= 1) {
    xmn = fminf(xmn, __shfl_xor(xmn, o));
    xmx = fmaxf(xmx, __shfl_xor(xmx, o));
    ymn = fminf(ymn, __shfl_xor(ymn, o));
    ymx = fmaxf(ymx, __shfl_xor(ymx, o));
  }

  __shared__ float ssum[kThreads / 32][6];
  __shared__ float smm[kThreads / 32][4];
  const int wave = tid >> 5, lane = tid & 31;
  if (lane == 0) {
#pragma unroll
    for (int k = 0; k < 6; ++k) ssum[wave][k] = sum[k];
    smm[wave][0] = xmn; smm[wave][1] = xmx; smm[wave][2] = ymn; smm[wave][3] = ymx;
  }
  __syncthreads();
  if (tid == 0) {
    float tot[6] = {0.f, 0.f, 0.f, 0.f, 0.f, 0.f};
    float mnx = __builtin_inff(), mxx = -__builtin_inff();
    float mny = __builtin_inff(), mxy = -__builtin_inff();
    for (int w = 0; w < kThreads / 32; ++w) {
#pragma unroll
      for (int k = 0; k < 6; ++k) tot[k] += ssum[w][k];
      mnx = fminf(mnx, smm[w][0]); mxx = fmaxf(mxx, smm[w][1]);
      mny = fminf(mny, smm[w][2]); mxy = fmaxf(mxy, smm[w][3]);
    }
    float* wsf = (float*)ws;
    atomicAdd(&wsf[WS_XSUM + b * 3 + 0], tot[0]);
    atomicAdd(&wsf[WS_XSUM + b * 3 + 1], tot[1]);
    atomicAdd(&wsf[WS_XSUM + b * 3 + 2], tot[2]);
    atomicAdd(&wsf[WS_YSUM + b * 3 + 0], tot[3]);
    atomicAdd(&wsf[WS_YSUM + b * 3 + 1], tot[4]);
    atomicAdd(&wsf[WS_YSUM + b * 3 + 2], tot[5]);
    atomicMin(&ws[WS_XMIN + b], enc_f(mnx));
    atomicMax(&ws[WS_XMAX + b], enc_f(mxx));
    atomicMin(&ws[WS_YMIN + b], enc_f(mny));
    atomicMax(&ws[WS_YMAX + b], enc_f(mxy));
  }
}

// Pass 2: recompute gray (L2-resident), bin into LDS histograms, flush with u32 atomics.
__global__ __launch_bounds__(kThreads) void cc_pass2(const float* __restrict__ x,
                                                     const float* __restrict__ y,
                                                     unsigned* __restrict__ ws) {
  __shared__ unsigned hx[kBins], hy[kBins];
  __shared__ float prm[4];
  const int blk = blockIdx.x;
  const int b   = blk / kBlocksPerImg;
  const int s   = blk % kBlocksPerImg;
  const int tid = threadIdx.x;

  if (tid < kBins) hx[tid] = 0u;
  else if (tid < 2 * kBins) hy[tid - kBins] = 0u;
  if (tid == 0) {
    float mn = dec_f(ws[WS_XMIN + b]);
    float mx = dec_f(ws[WS_XMAX + b]);
    prm[0] = mn; prm[1] = (mx > mn) ? 1.0f / (mx - mn) : 1.0f;
    mn = dec_f(ws[WS_YMIN + b]);
    mx = dec_f(ws[WS_YMAX + b]);
    prm[2] = mn; prm[3] = (mx > mn) ? 1.0f / (mx - mn) : 1.0f;
  }
  __syncthreads();
  const float xmn = prm[0], xinv = prm[1], ymn = prm[2], yinv = prm[3];

  const float* xr = x + (size_t)b * 3 * kN;
  const float* xg = xr + kN;
  const float* xb = xg + kN;
  const float* yr = y + (size_t)b * 3 * kN;
  const float* yg = yr + kN;
  const float* yb = yg + kN;

  const int base = s * kPxPerBlock + tid;
#pragma unroll 4
  for (int it = 0; it < kP2Iters; ++it) {
    const int p = base + it * kThreads;
    float gx = grayf(xr[p], xg[p], xb[p]);
    float gy = grayf(yr[p], yg[p], yb[p]);
    int ix = (int)(((gx - xmn) * xinv) * 63.0f);
    int iy = (int)(((gy - ymn) * yinv) * 63.0f);
    ix = ix < 0 ? 0 : (ix > 63 ? 63 : ix);
    iy = iy < 0 ? 0 : (iy > 63 ? 63 : iy);
    atomicAdd(&hx[ix], 1u);
    atomicAdd(&hy[iy], 1u);
  }
  __syncthreads();
  if (tid < kBins)            atomicAdd(&ws[WS_XHIST + b * kBins + tid], hx[tid]);
  else if (tid < 2 * kBins)   atomicAdd(&ws[WS_YHIST + b * kBins + (tid - kBins)], hy[tid - kBins]);
}

__global__ __launch_bounds__(kThreads) void cc_finalize(const unsigned* __restrict__ ws,
                                                        float* __restrict__ out) {
  __shared__ float s_cb, s_kl;
  const float* wsf = (const float*)ws;
  const int tid = threadIdx.x;
  if (tid == 0) { s_cb = 0.0f; s_kl = 0.0f; }
  __syncthreads();

  // color balance: 16 images x 3 channels
  if (tid < kB * 3) {
    const int b = tid / 3, c = tid % 3;
    float xm[3], ym[3];
#pragma unroll
    for (int k = 0; k < 3; ++k) {
      xm[k] = wsf[WS_XSUM + b * 3 + k] * (1.0f / (float)kN);
      ym[k] = wsf[WS_YSUM + b * 3 + k] * (1.0f / (float)kN);
    }
    const float xbal = xm[c] / (xm[0] + xm[1] + xm[2] + kEps);
    const float ybal = ym[c] / (ym[0] + ym[1] + ym[2] + kEps);
    atomicAdd(&s_cb, fabsf(xbal - ybal));
  }

  // KL over 16 x 64 histogram entries
  float kl = 0.0f;
  for (int k = tid; k < kB * kBins; k += kThreads) {
    const int b = k / kBins, i = k % kBins;
    const bool vx = dec_f(ws[WS_XMAX + b]) > dec_f(ws[WS_XMIN + b]);
    const bool vy = dec_f(ws[WS_YMAX + b]) > dec_f(ws[WS_YMIN + b]);
    const float hxv = vx ? (float)ws[WS_XHIST + b * kBins + i] * (1.0f / (float)kN)
                         : (1.0f / (float)kBins);
    const float hyv = vy ? (float)ws[WS_YHIST + b * kBins + i] * (1.0f / (float)kN)
                         : (1.0f / (float)kBins);
    kl += (hyv > 0.0f ? hyv * logf(hyv) : 0.0f) - hyv * logf(hxv + kEps);
  }
  atomicAdd(&s_kl, kl);
  __syncthreads();
  if (tid == 0) out[0] = 10.0f * (s_cb * (1.0f / 48.0f) + s_kl * (1.0f / 16.0f));
}

extern "C" void kernel_launch(void* const* d_in, const int* in_sizes, int n_in,
                              void* d_out, int out_size, void* d_ws, size_t ws_size,
                              hipStream_t stream) {
  (void)in_sizes; (void)n_in; (void)out_size; (void)ws_size;
  const float* x = (const float*)d_in[0];
  const float* y = (const float*)d_in[1];
  float* out     = (float*)d_out;
  unsigned* ws   = (unsigned*)d_ws;

  cc_init<<<(WS_WORDS + kThreads - 1) / kThreads, kThreads, 0, stream>>>(ws);
  cc_pass1<<<kB * kBlocksPerImg, kThreads, 0, stream>>>(x, y, ws);
  cc_pass2<<<kB * kBlocksPerImg, kThreads, 0, stream>>>(x, y, ws);
  cc_finalize<<<1, kThreads, 0, stream>>>(ws, out);
}